// SpdScaGr_29987461660842
// MI455X (gfx1250) — compile-verified
//
#include <hip/hip_runtime.h>
#include <hip/hip_bf16.h>
#include <math.h>

// Problem constants (match reference)
#define BATCH 256
#define SEQ   32
#define D_SPD 64
#define N_GR  128
#define P_GR  32
#define Q_GR  96

typedef float v2f __attribute__((ext_vector_type(2)));
typedef float v8f __attribute__((ext_vector_type(8)));

// Workspace layout (float offsets)
#define WS_TR    0        // 64x64 sym(q_tr_spd)
#define WS_EBIAS 4096     // 64x64 expm(sym(q_bias_spd))
#define WS_YB    8192     // 128x128 cayley(tangent(q_bias_gr))
#define WS_DSPD  24576    // 256 per-batch SPD distances
#define WS_DGR   24832    // 256 per-batch GR distances

#define WMMA_F32(a, b, c) \
  __builtin_amdgcn_wmma_f32_16x16x4_f32(false, (a), false, (b), (short)0, (c), false, false)

// ---------------------------------------------------------------------------
// Generic WMMA f32 GEMM: C[MxN] = alpha * op(A)[MxK] * op(B)[KxN] (+C if accum)
// Register-blocked: 2x2 tiles/wave for large GEMMs (A,B frags reused across 2
// WMMAs each -> LDS traffic per WMMA halved), 1x2 for 64x64 (keeps all 8 waves
// busy, A-frag reuse). Tile loops are uniform per wave -> EXEC all-ones at WMMA.
// Fragment layouts per CDNA5 ISA 7.12.2:
//   A 16x4: lane(m=l&15,h=l>>4): {A(m,k+2h), A(m,k+2h+1)}
//   B 4x16: {B(k+2h,n), B(k+2h+1,n)}
//   C 16x16: vgpr r holds C(r+8h, n)
// ---------------------------------------------------------------------------
__device__ inline void wgemm(const float* A, int lda, int tA,
                             const float* B, int ldb, int tB,
                             float* C, int ldc,
                             int M, int N, int K, float alpha, int accum)
{
  const int lane = threadIdx.x & 31;
  const int wave = threadIdx.x >> 5;
  const int nw   = blockDim.x >> 5;
  const int mi   = lane & 15;
  const int hi   = lane >> 4;
  const int mt = M >> 4, nt = N >> 4;

  auto loadA = [&](int row, int ka) -> v2f {
    v2f a;
    if (tA) { a.x = A[(ka + 0) * lda + row + mi]; a.y = A[(ka + 1) * lda + row + mi]; }
    else    { a.x = A[(row + mi) * lda + ka + 0]; a.y = A[(row + mi) * lda + ka + 1]; }
    return a;
  };
  auto loadB = [&](int col, int ka) -> v2f {
    v2f b;
    if (tB) { b.x = B[(col + mi) * ldb + ka + 0]; b.y = B[(col + mi) * ldb + ka + 1]; }
    else    { b.x = B[(ka + 0) * ldb + col + mi]; b.y = B[(ka + 1) * ldb + col + mi]; }
    return b;
  };
  auto store = [&](const v8f& acc, int ti, int tj) {
#pragma unroll
    for (int r = 0; r < 8; ++r) {
      const int ci = (ti + r + 8 * hi) * ldc + tj + mi;
      float v = alpha * acc[r];
      C[ci] = accum ? (C[ci] + v) : v;
    }
  };

  if (((mt & 1) == 0) && ((nt & 1) == 0) && (((mt * nt) >> 2) >= nw)) {
    // ---- 2x2 blocking: wave owns a 32x32 macro-tile ----
    const int bm = mt >> 1, bn = nt >> 1;
    for (int t = wave; t < bm * bn; t += nw) {
      const int ti = (t / bn) << 5;
      const int tj = (t % bn) << 5;
      v8f a00 = {}, a01 = {}, a10 = {}, a11 = {};
      for (int k = 0; k < K; k += 4) {
        const int ka = k + 2 * hi;
        v2f af0 = loadA(ti, ka), af1 = loadA(ti + 16, ka);
        v2f bf0 = loadB(tj, ka), bf1 = loadB(tj + 16, ka);
        a00 = WMMA_F32(af0, bf0, a00);
        a01 = WMMA_F32(af0, bf1, a01);
        a10 = WMMA_F32(af1, bf0, a10);
        a11 = WMMA_F32(af1, bf1, a11);
      }
      store(a00, ti, tj);       store(a01, ti, tj + 16);
      store(a10, ti + 16, tj);  store(a11, ti + 16, tj + 16);
    }
  } else if (((nt & 1) == 0) && (((mt * nt) >> 1) >= nw)) {
    // ---- 1x2 blocking: wave owns a 16x32 macro-tile (A-frag reused) ----
    const int bn = nt >> 1;
    for (int t = wave; t < mt * bn; t += nw) {
      const int ti = (t / bn) << 4;
      const int tj = (t % bn) << 5;
      v8f a0 = {}, a1 = {};
      for (int k = 0; k < K; k += 4) {
        const int ka = k + 2 * hi;
        v2f af = loadA(ti, ka);
        v2f bf0 = loadB(tj, ka), bf1 = loadB(tj + 16, ka);
        a0 = WMMA_F32(af, bf0, a0);
        a1 = WMMA_F32(af, bf1, a1);
      }
      store(a0, ti, tj);
      store(a1, ti, tj + 16);
    }
  } else {
    // ---- 1x1 fallback ----
    for (int t = wave; t < mt * nt; t += nw) {
      const int ti = (t / nt) << 4;
      const int tj = (t % nt) << 4;
      v8f acc = {};
      for (int k = 0; k < K; k += 4) {
        const int ka = k + 2 * hi;
        acc = WMMA_F32(loadA(ti, ka), loadB(tj, ka), acc);
      }
      store(acc, ti, tj);
    }
  }
  __syncthreads();
}

// ---------------------------------------------------------------------------
// Elementwise helpers (all end with a block barrier; all calls are uniform)
// ---------------------------------------------------------------------------
__device__ inline void ew_ai_bx(float* dst, const float* src, float a, float b, int D) {
  const int n = D * D;
  for (int i = threadIdx.x; i < n; i += blockDim.x) {
    float v = b * src[i];
    if ((i % (D + 1)) == 0) v += a;      // dst = a*I + b*src
    dst[i] = v;
  }
  __syncthreads();
}
__device__ inline void ew_setI(float* dst, int D) {
  const int n = D * D;
  for (int i = threadIdx.x; i < n; i += blockDim.x)
    dst[i] = ((i % (D + 1)) == 0) ? 1.0f : 0.0f;
  __syncthreads();
}
__device__ inline void ew_copy(float* dst, const float* src, int n) {
  for (int i = threadIdx.x; i < n; i += blockDim.x) dst[i] = src[i];
  __syncthreads();
}
__device__ inline void ew_mul(float* dst, const float* src, int n) {
  for (int i = threadIdx.x; i < n; i += blockDim.x) dst[i] *= src[i];
  __syncthreads();
}
__device__ inline void ew_sym(float* dst, const float* src, int D) {
  const int n = D * D;
  for (int i = threadIdx.x; i < n; i += blockDim.x) {
    int r = i / D, c = i % D;
    dst[i] = 0.5f * (src[r * D + c] + src[c * D + r]);
  }
  __syncthreads();
}
__device__ inline void ew_trans_scale(float* dst, const float* src, float s, int D) {
  const int n = D * D;
  for (int i = threadIdx.x; i < n; i += blockDim.x) {
    int r = i / D, c = i % D;
    dst[i] = s * src[c * D + r];
  }
  __syncthreads();
}

// Block-wide sum via LDS float atomic (ds_add_f32)
__device__ inline float block_sum(float v, float* slot) {
  __syncthreads();
  if (threadIdx.x == 0) *slot = 0.0f;
  __syncthreads();
  atomicAdd(slot, v);
  __syncthreads();
  float r = *slot;
  __syncthreads();
  return r;
}
__device__ inline float block_trace(const float* A, int D, float* slot) {
  float p = 0.0f;
  for (int i = threadIdx.x; i < D; i += blockDim.x) p += A[i * D + i];
  return block_sum(p, slot);
}
__device__ inline float block_fro2(const float* A, int n, float* slot) {
  float p = 0.0f;
  for (int i = threadIdx.x; i < n; i += blockDim.x) { float x = A[i]; p += x * x; }
  return block_sum(p, slot);
}

// dst = scale * sym(64x64 e)
__device__ inline void load_sym_tok(float* dst, const float* e, float scale) {
  for (int i = threadIdx.x; i < 4096; i += blockDim.x) {
    int r = i >> 6, c = i & 63;
    dst[i] = scale * 0.5f * (e[i] + e[c * 64 + r]);
  }
  __syncthreads();
}

// ---------------------------------------------------------------------------
// Matmul-only matrix functions (64x64)
// ---------------------------------------------------------------------------
// E = expm(A) via Horner-Taylor: E=I; for j=terms..1: E = I + (A*E)/j
__device__ inline void expm_taylor64(const float* A, float* E, float* T, int terms) {
  ew_setI(E, 64);
  for (int j = terms; j >= 1; --j) {
    wgemm(A, 64, 0, E, 64, 0, T, 64, 64, 64, 64, 1.0f / (float)j, 0);
    ew_ai_bx(E, T, 1.0f, 1.0f, 64);
  }
}

// Coupled Newton-Schulz: Y->A^{1/2}, Z->A^{-1/2} (A pre-normalized near I)
__device__ inline void ns_sqrt_iter64(float* Y, float* Z, float* T, float* W, int iters) {
  for (int it = 0; it < iters; ++it) {
    wgemm(Z, 64, 0, Y, 64, 0, T, 64, 64, 64, 64, 1.0f, 0);
    ew_ai_bx(W, T, 1.5f, -0.5f, 64);                 // W = (3I - Z*Y)/2
    wgemm(Y, 64, 0, W, 64, 0, T, 64, 64, 64, 64, 1.0f, 0);
    ew_copy(Y, T, 4096);
    wgemm(W, 64, 0, Z, 64, 0, T, 64, 64, 64, 64, 1.0f, 0);
    ew_copy(Z, T, 4096);
  }
}

// Trace-normalized pair: Y = sqrtm(A), Z = A^{-1/2}
__device__ inline void ns_pair64(const float* A, float* Y, float* Z,
                                 float* T, float* W, float* slot, int iters) {
  float tr = block_trace(A, 64, slot);
  float c  = 64.0f / tr;
  ew_ai_bx(Y, A, 0.0f, c, 64);
  ew_setI(Z, 64);
  ns_sqrt_iter64(Y, Z, T, W, iters);
  float sc = sqrtf(fabsf(c));
  ew_ai_bx(Y, Y, 0.0f, 1.0f / sc, 64);
  ew_ai_bx(Z, Z, 0.0f, sc, 64);
}

// L = logm(A) via inverse scaling-and-squaring (3 sqrtm + 4-term Gregory). A destroyed.
__device__ inline void logm64(float* A, float* L, float* Y, float* Z,
                              float* T, float* W, float* slot) {
  float tr = block_trace(A, 64, slot);
  float c  = 64.0f / tr;
  ew_ai_bx(A, A, 0.0f, c, 64);
  for (int s = 0; s < 3; ++s) {
    ew_copy(Y, A, 4096);
    ew_setI(Z, 64);
    ns_sqrt_iter64(Y, Z, T, W, 4);
    ew_copy(A, Y, 4096);
  }
  ew_ai_bx(A, A, -1.0f, 1.0f, 64);                    // E = A - I
  ew_ai_bx(Y, A, 1.0f / 3.0f, -0.25f, 64);            // P = I/3 - E/4
  wgemm(A, 64, 0, Y, 64, 0, T, 64, 64, 64, 64, 1.0f, 0);
  ew_ai_bx(Y, T, -0.5f, 1.0f, 64);                    // P = E*P - I/2
  wgemm(A, 64, 0, Y, 64, 0, T, 64, 64, 64, 64, 1.0f, 0);
  ew_ai_bx(Y, T, 1.0f, 1.0f, 64);                     // P = E*P + I
  wgemm(A, 64, 0, Y, 64, 0, L, 64, 64, 64, 64, 8.0f, 0);  // L = 8*E*P
  ew_ai_bx(L, L, -logf(fabsf(c)), 1.0f, 64);          // undo trace normalization
}

// y = expm(x_{T-1}); for t=T-2..0: y = expm(x_t/2) y expm(x_t/2)
__device__ inline void spd_accumulate(const int* ids, const float* emb,
                                      float* Y, float* X, float* S, float* T, int b) {
  const float* e = emb + (size_t)ids[b * SEQ + (SEQ - 1)] * 4096;
  load_sym_tok(X, e, 1.0f);
  expm_taylor64(X, Y, T, 6);
  for (int t = SEQ - 2; t >= 0; --t) {
    const float* et = emb + (size_t)ids[b * SEQ + t] * 4096;
    __builtin_prefetch(et, 0, 0);
    load_sym_tok(X, et, 0.5f);
    expm_taylor64(X, S, T, 6);
    wgemm(S, 64, 0, Y, 64, 0, T, 64, 64, 64, 64, 1.0f, 0);
    wgemm(T, 64, 0, S, 64, 0, Y, 64, 64, 64, 64, 1.0f, 0);
  }
}

// ---------------------------------------------------------------------------
// Grassmann helpers (128x128)
// ---------------------------------------------------------------------------
__device__ inline void gr_build_skew(float* dst, const float* e, const float* tr) {
  for (int i = threadIdx.x; i < N_GR * N_GR; i += blockDim.x) {
    int r = i >> 7, c = i & 127;
    float v = 0.0f;
    if (r < P_GR && c >= P_GR) {
      int j = c - P_GR;
      float s = tr ? tr[r * Q_GR + j] : 1.0f;
      v = s * e[r * Q_GR + j];
    } else if (c < P_GR && r >= P_GR) {
      int j = r - P_GR;
      float s = tr ? tr[c * Q_GR + j] : 1.0f;
      v = -s * e[c * Q_GR + j];
    }
    dst[i] = v;
  }
  __syncthreads();
}

// OUT = cayley(X) ~ I - 2X + 2X^2 - 2X^3 + 2X^4 (Horner, 3 GEMMs); ||X|| small
__device__ inline void cayley128(const float* X, float* OUT, float* SCR) {
  ew_ai_bx(OUT, X, 1.0f, -1.0f, 128);
  wgemm(X, 128, 0, OUT, 128, 0, SCR, 128, 128, 128, 128, 1.0f, 0);
  ew_ai_bx(OUT, SCR, 1.0f, -1.0f, 128);
  wgemm(X, 128, 0, OUT, 128, 0, SCR, 128, 128, 128, 128, 1.0f, 0);
  ew_ai_bx(OUT, SCR, 1.0f, -1.0f, 128);
  wgemm(X, 128, 0, OUT, 128, 0, SCR, 128, 128, 128, 128, 1.0f, 0);
  ew_ai_bx(OUT, SCR, 1.0f, -2.0f, 128);
}

__device__ inline void gr_scan(const int* ids, const float* emb, const float* tr,
                               float* Z, float* X, float* Y, float* T, int b) {
  gr_build_skew(X, emb + (size_t)ids[b * SEQ] * (P_GR * Q_GR), tr);
  cayley128(X, Z, T);
  for (int t = 1; t < SEQ; ++t) {
    const float* et = emb + (size_t)ids[b * SEQ + t] * (P_GR * Q_GR);
    __builtin_prefetch(et, 0, 0);
    gr_build_skew(X, et, tr);
    cayley128(X, Y, T);
    wgemm(Z, 128, 0, Y, 128, 0, T, 128, 128, 128, 128, 1.0f, 0);
    ew_copy(Z, T, 16384);
  }
}

// ---------------------------------------------------------------------------
// Kernel 0: one-time constants (trS, expm(biasS), cayley(bias_gr) via NS inverse)
// ---------------------------------------------------------------------------
__global__ void __launch_bounds__(256)
spd_precompute_kernel(const float* __restrict__ q_tr_spd,
                      const float* __restrict__ q_bias_spd,
                      const float* __restrict__ q_bias_gr,
                      float* __restrict__ ws)
{
  extern __shared__ float sm[];
  float* S0 = sm; float* S1 = sm + 4096; float* S2 = sm + 8192;

  // trS = sym(q_tr_spd)
  for (int i = threadIdx.x; i < 4096; i += blockDim.x) {
    int r = i >> 6, c = i & 63;
    ws[WS_TR + i] = 0.5f * (q_tr_spd[i] + q_tr_spd[c * 64 + r]);
  }
  // Ebias = expm(sym(q_bias_spd)): scale 1/16, Taylor-8, square 4x
  load_sym_tok(S0, q_bias_spd, 1.0f / 16.0f);
  expm_taylor64(S0, S1, S2, 8);
  wgemm(S1, 64, 0, S1, 64, 0, S2, 64, 64, 64, 64, 1.0f, 0);
  wgemm(S2, 64, 0, S2, 64, 0, S1, 64, 64, 64, 64, 1.0f, 0);
  wgemm(S1, 64, 0, S1, 64, 0, S2, 64, 64, 64, 64, 1.0f, 0);
  wgemm(S2, 64, 0, S2, 64, 0, S1, 64, 64, 64, 64, 1.0f, 0);
  for (int i = threadIdx.x; i < 4096; i += blockDim.x) ws[WS_EBIAS + i] = S1[i];
  __syncthreads();

  // Yb = cayley(tangent(q_bias_gr)) = (I-X)(I+X)^{-1}, NS inverse (25 iters)
  float* G0 = sm;            float* G1 = sm + 16384;
  float* G2 = sm + 32768;    float* G3 = sm + 49152;
  float* G4 = sm + 65536;
  gr_build_skew(G0, q_bias_gr, nullptr);
  ew_ai_bx(G1, G0, 1.0f, 1.0f, 128);                 // M = I + X
  float f2 = block_fro2(G1, 16384, &G4[0]);
  ew_trans_scale(G2, G1, 1.0f / f2, 128);            // V0 = M^T / ||M||_F^2
  for (int it = 0; it < 25; ++it) {
    wgemm(G1, 128, 0, G2, 128, 0, G3, 128, 128, 128, 128, 1.0f, 0);
    ew_ai_bx(G3, G3, 2.0f, -1.0f, 128);              // 2I - M*V
    wgemm(G2, 128, 0, G3, 128, 0, G4, 128, 128, 128, 128, 1.0f, 0);
    ew_copy(G2, G4, 16384);                          // V = V*(2I - M*V)
  }
  ew_ai_bx(G1, G0, 1.0f, -1.0f, 128);                // I - X
  wgemm(G1, 128, 0, G2, 128, 0, G3, 128, 128, 128, 128, 1.0f, 0);
  for (int i = threadIdx.x; i < 16384; i += blockDim.x) ws[WS_YB + i] = G3[i];
}

// ---------------------------------------------------------------------------
// Kernel 1: SPD branch, one workgroup per batch element
// ---------------------------------------------------------------------------
__global__ void __launch_bounds__(256)
spd_kernel(const int* __restrict__ qids, const int* __restrict__ aids,
           const float* __restrict__ q_emb, const float* __restrict__ a_emb,
           const float* __restrict__ ws, float* __restrict__ dist)
{
  extern __shared__ float sm[];
  float* B0 = sm;             float* B1 = sm + 4096;
  float* B2 = sm + 2 * 4096;  float* B3 = sm + 3 * 4096;
  float* B4 = sm + 4 * 4096;  float* B5 = sm + 5 * 4096;
  float* B6 = sm + 6 * 4096;  float* slot = sm + 7 * 4096;
  const int b = blockIdx.x;

  spd_accumulate(qids, q_emb, B1, B0, B2, B3, b);    // B1 = Q
  spd_accumulate(aids, a_emb, B4, B0, B2, B3, b);    // B4 = Aacc

  logm64(B1, B0, B2, B3, B5, B6, slot);              // B0 = logm(Q)
  ew_mul(B0, ws + WS_TR, 4096);                      // tangent scaling: tr .* log
  ew_ai_bx(B0, B0, 0.0f, 0.25f, 64);                 // /4 for scaling-squaring
  expm_taylor64(B0, B1, B2, 6);
  wgemm(B1, 64, 0, B1, 64, 0, B2, 64, 64, 64, 64, 1.0f, 0);
  wgemm(B2, 64, 0, B2, 64, 0, B1, 64, 64, 64, 64, 1.0f, 0);   // B1 = Q2
  ns_pair64(B1, B2, B3, B5, B6, slot, 5);            // B2 = sqrtm(Q2)
  wgemm(B2, 64, 0, ws + WS_EBIAS, 64, 0, B3, 64, 64, 64, 64, 1.0f, 0);
  wgemm(B3, 64, 0, B2, 64, 0, B1, 64, 64, 64, 64, 1.0f, 0);   // B1 = sq*Ebias*sq
  ns_pair64(B4, B2, B3, B5, B6, slot, 5);            // B3 = Aacc^{-1/2}
  wgemm(B3, 64, 0, B1, 64, 0, B5, 64, 64, 64, 64, 1.0f, 0);
  wgemm(B5, 64, 0, B3, 64, 0, B6, 64, 64, 64, 64, 1.0f, 0);
  ew_sym(B1, B6, 64);                                // M = sym(isa*Q*isa)
  logm64(B1, B0, B2, B3, B5, B6, slot);              // B0 = logm(M)
  float p = 0.0f;
  for (int i = threadIdx.x; i < 4096; i += blockDim.x) { float x = B0[i]; p += x * x; }
  float s = block_sum(p, slot);
  if (threadIdx.x == 0) dist[b] = sqrtf(s);
}

// ---------------------------------------------------------------------------
// Kernel 2: Grassmann branch, one workgroup per batch element
// ---------------------------------------------------------------------------
__global__ void __launch_bounds__(256)
gr_kernel(const int* __restrict__ qids, const int* __restrict__ aids,
          const float* __restrict__ qe, const float* __restrict__ ae,
          const float* __restrict__ trg, const float* __restrict__ ws,
          float* __restrict__ dist)
{
  extern __shared__ float sm[];
  float* G0 = sm;            float* G1 = sm + 16384;
  float* G2 = sm + 32768;    float* G3 = sm + 49152;
  float* G4 = sm + 65536;
  const int b = blockIdx.x;
  const float* Yb = ws + WS_YB;

  // Question: Z = prod cayley(x_t); q_gr = Yb * (Z_p Z_p^T) * Yb^T
  gr_scan(qids, qe, trg, G0, G1, G2, G3, b);
  wgemm(G0, 128, 0, G0, 128, 1, G2, 128, 128, 128, 32, 1.0f, 0);   // Z_p Z_p^T
  wgemm(Yb, 128, 0, G2, 128, 0, G3, 128, 128, 128, 128, 1.0f, 0);
  wgemm(G3, 128, 0, Yb, 128, 1, G4, 128, 128, 128, 128, 1.0f, 0);  // G4 = q_gr

  // Answer: a_gr = Z_p Z_p^T
  gr_scan(aids, ae, nullptr, G0, G1, G2, G3, b);
  wgemm(G0, 128, 0, G0, 128, 1, G2, 128, 128, 128, 32, 1.0f, 0);   // G2 = a_gr

  float p = 0.0f;
  for (int i = threadIdx.x; i < 16384; i += blockDim.x) {
    float d = G4[i] - G2[i];
    p += d * d;
  }
  float s = block_sum(p, &G0[0]);
  if (threadIdx.x == 0) dist[b] = sqrtf(s);
}

// ---------------------------------------------------------------------------
// Kernel 3: combine
// ---------------------------------------------------------------------------
__global__ void combine_kernel(const float* __restrict__ dspd,
                               const float* __restrict__ dgr,
                               const float* __restrict__ wf,
                               const float* __restrict__ wb,
                               float* __restrict__ out)
{
  int i = blockIdx.x * blockDim.x + threadIdx.x;
  if (i < BATCH) out[i] = -wf[0] * (dspd[i] + dgr[i]) + wb[0];
}

// ---------------------------------------------------------------------------
extern "C" void kernel_launch(void* const* d_in, const int* in_sizes, int n_in,
                              void* d_out, int out_size, void* d_ws, size_t ws_size,
                              hipStream_t stream)
{
  (void)in_sizes; (void)n_in; (void)out_size; (void)ws_size;
  const int*   qids      = (const int*)d_in[0];
  const int*   aids      = (const int*)d_in[1];
  const float* q_emb_spd = (const float*)d_in[2];
  const float* a_emb_spd = (const float*)d_in[3];
  const float* q_tr_spd  = (const float*)d_in[4];
  const float* q_bias_spd= (const float*)d_in[5];
  const float* q_emb_gr  = (const float*)d_in[6];
  const float* a_emb_gr  = (const float*)d_in[7];
  const float* q_tr_gr   = (const float*)d_in[8];
  const float* q_bias_gr = (const float*)d_in[9];
  const float* wf        = (const float*)d_in[10];
  const float* wb        = (const float*)d_in[11];
  float* out = (float*)d_out;
  float* ws  = (float*)d_ws;

  const size_t sh_pre = (size_t)5 * 16384 * sizeof(float);           // 320 KB
  const size_t sh_spd = (size_t)(7 * 4096 + 32) * sizeof(float);     // ~112 KB
  const size_t sh_gr  = (size_t)5 * 16384 * sizeof(float);           // 320 KB

  spd_precompute_kernel<<<1, 256, sh_pre, stream>>>(q_tr_spd, q_bias_spd, q_bias_gr, ws);
  spd_kernel<<<BATCH, 256, sh_spd, stream>>>(qids, aids, q_emb_spd, a_emb_spd,
                                             ws, ws + WS_DSPD);
  gr_kernel<<<BATCH, 256, sh_gr, stream>>>(qids, aids, q_emb_gr, a_emb_gr,
                                           q_tr_gr, ws, ws + WS_DGR);
  combine_kernel<<<1, 256, 0, stream>>>(ws + WS_DSPD, ws + WS_DGR, wf, wb, out);
}